// classDistance_61409442398508
// MI455X (gfx1250) — compile-verified
//
#include <hip/hip_runtime.h>
#include <hip/hip_bf16.h>

// Problem constants (from reference)
#define B_ROWS 8192
#define DIMS   128
#define NCLS   6

// Grid plan for the WMMA class-sum kernel
#define NUM_WG        32
#define THREADS_PER_WG 256            // 8 waves (wave32)
#define WAVES_PER_WG  (THREADS_PER_WG / 32)
#define TOTAL_WAVES   (NUM_WG * WAVES_PER_WG)   // 256
#define ROWS_PER_WAVE (B_ROWS / TOTAL_WAVES)    // 32

// Workspace layout (floats):
//   [0, 8192)                      : inv_norm per row
//   [8192, 8192 + NUM_WG*768)      : per-workgroup partial M (6x128 each)
#define WS_INV_OFF  0
#define WS_PART_OFF B_ROWS
#define M_ELEMS     (NCLS * DIMS)   // 768

typedef float v2f __attribute__((ext_vector_type(2)));
typedef float v8f __attribute__((ext_vector_type(8)));

// ---------------------------------------------------------------------------
// Kernel 1: per-row inverse norms.  inv_norm[r] = 1 / max(||x_r||, 1e-5)
// (sqrt(EPS)=1e-5 guard; for this input norms ~ sqrt(128), clamp never fires)
// ---------------------------------------------------------------------------
__global__ __launch_bounds__(256) void k_norms(const float* __restrict__ latent,
                                               float* __restrict__ ws) {
    int t = blockIdx.x * blockDim.x + threadIdx.x;
    if (t >= B_ROWS) return;
    const float4* row = (const float4*)(latent + (size_t)t * DIMS);
    float ss = 0.0f;
#pragma unroll
    for (int i = 0; i < DIMS / 4; ++i) {
        float4 v = row[i];
        ss += v.x * v.x + v.y * v.y + v.z * v.z + v.w * v.w;
    }
    float n = sqrtf(ss);
    ws[WS_INV_OFF + t] = 1.0f / fmaxf(n, 1e-5f);
}

// ---------------------------------------------------------------------------
// Kernel 2: M = onehot(label)^T @ (latent * inv_norm)  via V_WMMA_F32_16X16X4_F32
//   A (16x4): A[c][k] = (label[row_k]==c) ? inv_norm[row_k] : 0   (norm folded in)
//   B (4x16): B[k][n] = latent[row_k][16*tile + n]
//   D (16x16) accumulates class-sums for one 16-wide column tile; 8 tiles = 128 dims.
// Per-wave partials go to a private LDS slab (no atomics), reduced in fixed order.
// ---------------------------------------------------------------------------
__global__ __launch_bounds__(THREADS_PER_WG) void k_classsum(
        const float* __restrict__ latent,
        const int*   __restrict__ label,
        float*       __restrict__ ws) {
    __shared__ float ldsM[WAVES_PER_WG * M_ELEMS];   // 8 * 768 floats = 24 KB

    const float* invn = ws + WS_INV_OFF;

    const int lane   = threadIdx.x & 31;
    const int j      = lane & 15;        // A: class row (M) / B: column (N)
    const int half   = lane >> 4;        // K split: 0 -> K={0,1}, 1 -> K={2,3}
    const int waveIn = threadIdx.x >> 5; // wave index within WG
    const int waveId = (blockIdx.x * blockDim.x + threadIdx.x) >> 5;
    const int rowBase = waveId * ROWS_PER_WAVE;

    v8f acc[8];
#pragma unroll
    for (int n = 0; n < 8; ++n)
#pragma unroll
        for (int e = 0; e < 8; ++e) acc[n][e] = 0.0f;

    for (int k0 = 0; k0 < ROWS_PER_WAVE; k0 += 4) {
        const int rA = rowBase + k0 + 2 * half;   // VGPR .x -> K = 2*half
        const int rB = rA + 1;                    // VGPR .y -> K = 2*half + 1
        const float ia = invn[rA];
        const float ib = invn[rB];
        const int la = label[rA];
        const int lb = label[rB];

        v2f a;
        a.x = (la == j) ? ia : 0.0f;
        a.y = (lb == j) ? ib : 0.0f;

        const float* prA = latent + (size_t)rA * DIMS;
        const float* prB = latent + (size_t)rB * DIMS;
#pragma unroll
        for (int n = 0; n < 8; ++n) {
            v2f b;
            b.x = prA[16 * n + j];   // lanes 0-15 of each half: 64B coalesced
            b.y = prB[16 * n + j];
            // (neg_a, A, neg_b, B, c_mod, C, reuse_a, reuse_b)
            acc[n] = __builtin_amdgcn_wmma_f32_16x16x4_f32(
                false, a, false, b, (short)0, acc[n], false, false);
        }
    }

    // D layout: lane j<16, VGPR r -> D[r][16n+j]; lanes>=16 hold rows 8..15 (all 0).
    // Rows 6..15 are structurally zero (one-hot pads) -> only r<6 of the low half.
    if (half == 0) {
#pragma unroll
        for (int n = 0; n < 8; ++n)
#pragma unroll
            for (int r = 0; r < NCLS; ++r)
                ldsM[waveIn * M_ELEMS + r * DIMS + 16 * n + j] = acc[n][r];
    }
    __syncthreads();

    // Fixed-order cross-wave reduction -> one deterministic partial per WG.
    for (int t = threadIdx.x; t < M_ELEMS; t += blockDim.x) {
        float s = 0.0f;
#pragma unroll
        for (int w = 0; w < WAVES_PER_WG; ++w) s += ldsM[w * M_ELEMS + t];
        ws[WS_PART_OFF + blockIdx.x * M_ELEMS + t] = s;
    }
}

// ---------------------------------------------------------------------------
// Kernel 3: reduce WG partials (fixed order), count labels, form the 15
// upper-triangle entries of P = M M^T, divide by max(n_i^2 - n_i, 1), mean.
// ---------------------------------------------------------------------------
__global__ __launch_bounds__(256) void k_final(const int*   __restrict__ label,
                                               const float* __restrict__ ws,
                                               float*       __restrict__ out) {
    __shared__ float Msh[M_ELEMS];
    __shared__ int   cnt[NCLS];
    __shared__ float pairv[16];

    if (threadIdx.x < NCLS) cnt[threadIdx.x] = 0;
    if (threadIdx.x < 16)   pairv[threadIdx.x] = 0.0f;

    for (int t = threadIdx.x; t < M_ELEMS; t += blockDim.x) {
        float s = 0.0f;
        for (int g = 0; g < NUM_WG; ++g)        // fixed order -> deterministic
            s += ws[WS_PART_OFF + g * M_ELEMS + t];
        Msh[t] = s;
    }
    __syncthreads();

    // Integer label counts (LDS atomics; exact & order-independent).
    for (int t = threadIdx.x; t < B_ROWS; t += blockDim.x) {
        int l = label[t];
        if (l >= 0 && l < NCLS) atomicAdd(&cnt[l], 1);
    }
    __syncthreads();

    if (threadIdx.x < 15) {
        int t = threadIdx.x, pi = 0, pj = 1, c = 0;
        for (int a = 0; a < NCLS; ++a)
            for (int b = a + 1; b < NCLS; ++b) {
                if (c == t) { pi = a; pj = b; }
                ++c;
            }
        float dot = 0.0f;
        for (int d = 0; d < DIMS; ++d)
            dot += Msh[pi * DIMS + d] * Msh[pj * DIMS + d];
        float ci = (float)cnt[pi];
        float denom = fmaxf(ci * ci - ci, 1.0f);
        pairv[t] = dot / denom;
    }
    __syncthreads();

    if (threadIdx.x == 0) {
        float s = 0.0f;
        for (int t = 0; t < 15; ++t) s += pairv[t];
        out[0] = s / 15.0f;
    }
}

// ---------------------------------------------------------------------------
extern "C" void kernel_launch(void* const* d_in, const int* in_sizes, int n_in,
                              void* d_out, int out_size, void* d_ws, size_t ws_size,
                              hipStream_t stream) {
    const float* latent = (const float*)d_in[0];
    const int*   label  = (const int*)d_in[1];
    float*       ws     = (float*)d_ws;
    float*       out    = (float*)d_out;

    k_norms<<<B_ROWS / 256, 256, 0, stream>>>(latent, ws);
    k_classsum<<<NUM_WG, THREADS_PER_WG, 0, stream>>>(latent, label, ws);
    k_final<<<1, 256, 0, stream>>>(label, ws, out);
}